// Dependency_GCN_32873679683801
// MI455X (gfx1250) — compile-verified
//
#include <hip/hip_runtime.h>

// Problem constants (match reference)
#define N_TOK  8192
#define D_DIM  128
#define RBASE  40
#define R_TOT  80            // forward + reversed labels
#define E_FWD  N_TOK
#define E_ALL  (2 * N_TOK)
#define LDS_STRIDE 132       // 128 + 4 pad -> conflict-free 16-row fragment reads
#define EDGE_CHUNKS 4        // blocks per relation
#define SELF_MTILES 4        // 16-row m-tiles per block (one per wave)

typedef __attribute__((ext_vector_type(2))) float v2f;
typedef __attribute__((ext_vector_type(8))) float v8f;

// ---- gfx1250 async global->LDS staging (ASYNCcnt path, ISA 08 §4) ----------
// VDST = per-lane LDS byte offset VGPR, VADDR = 64-bit global address.
__device__ __forceinline__ void async_load_b128(const float* gsrc, const float* lds_dst) {
    unsigned off = (unsigned)(size_t)lds_dst;   // flat LDS addr low 32 bits = LDS offset
    asm volatile("global_load_async_to_lds_b128 %0, %1, off"
                 :: "v"(off), "v"(gsrc) : "memory");
}
__device__ __forceinline__ void wait_async0() {
    asm volatile("s_wait_asynccnt 0x0" ::: "memory");
}

// ---------------- binning kernels ----------------

__global__ void zero_counts_k(int* counts) {
    int i = threadIdx.x;
    if (i < R_TOT) counts[i] = 0;
}

__device__ __forceinline__ int edge_rel(const int* rel, int e) {
    return (e < E_FWD) ? rel[e] : (rel[e - E_FWD] + RBASE);
}

__global__ void count_edges_k(const int* __restrict__ rel, int* counts) {
    int e = blockIdx.x * blockDim.x + threadIdx.x;
    if (e >= E_ALL) return;
    atomicAdd(&counts[edge_rel(rel, e)], 1);
}

__global__ void scan_offsets_k(const int* __restrict__ counts,
                               int* __restrict__ offsets,
                               int* __restrict__ cursors) {
    if (threadIdx.x == 0) {
        int acc = 0;
        for (int r = 0; r < R_TOT; ++r) {
            offsets[r] = acc;
            cursors[r] = acc;
            acc += counts[r];
        }
        offsets[R_TOT] = acc;   // == E_ALL
    }
}

__global__ void scatter_edges_k(const int* __restrict__ rel,
                                int* __restrict__ cursors,
                                int* __restrict__ perm) {
    int e = blockIdx.x * blockDim.x + threadIdx.x;
    if (e >= E_ALL) return;
    int r = edge_rel(rel, e);
    int pos = atomicAdd(&cursors[r], 1);
    perm[pos] = e;
}

// ---------------- self transform: out = X @ W_self^T + b_self ----------------
// 4 waves/block. Full W_self cached once in LDS (async), each wave owns one
// 16-row m-tile and runs 8 n-tiles x 32 x V_WMMA_F32_16X16X4_F32.

__global__ __launch_bounds__(128)
void self_gemm_k(const float* __restrict__ x,
                 const float* __restrict__ Ws,
                 const float* __restrict__ bs,
                 float* __restrict__ out) {
    extern __shared__ __align__(16) float smem[];
    float* lw = smem;                                   // 128 * LDS_STRIDE
    const int tid    = threadIdx.x;
    const int wave   = tid >> 5;
    const int lane   = tid & 31;
    const int lane16 = lane & 15;
    const int half   = lane >> 4;                       // K{0,1}/M{0..7} vs K{2,3}/M{8..15}
    float* lx = smem + 128 * LDS_STRIDE + wave * (16 * LDS_STRIDE);   // per-wave x tile

    // stage full W_self into LDS: wave w stages rows [32w, 32w+32)
#pragma unroll 1
    for (int j = 0; j < 32; ++j) {
        int row = wave * 32 + j;
        async_load_b128(Ws + row * 128 + lane * 4, &lw[row * LDS_STRIDE + lane * 4]);
    }
    // stage this wave's 16 x rows
    const int mbase = (blockIdx.x * SELF_MTILES + wave) * 16;
#pragma unroll 1
    for (int j = 0; j < 16; ++j) {
        async_load_b128(x + (size_t)(mbase + j) * 128 + lane * 4,
                        &lx[j * LDS_STRIDE + lane * 4]);
    }
    wait_async0();          // this wave's async copies landed in LDS
    __syncthreads();        // all waves' W rows visible

    const float* ax = &lx[lane16 * LDS_STRIDE + half * 2];
#pragma unroll 1
    for (int nt = 0; nt < 8; ++nt) {
        const float* bw = &lw[(nt * 16 + lane16) * LDS_STRIDE + half * 2];
        v8f acc = {};
#pragma unroll
        for (int kk = 0; kk < 32; ++kk) {
            v2f a = *(const v2f*)(ax + kk * 4);   // ds_load_b64, conflict-free
            v2f b = *(const v2f*)(bw + kk * 4);
            acc = __builtin_amdgcn_wmma_f32_16x16x4_f32(
                false, a, false, b, (short)0, acc, false, false);
        }
        float bias = bs[nt * 16 + lane16];
#pragma unroll
        for (int p = 0; p < 8; ++p) {
            int m = p + 8 * half;                 // C/D layout: M = p + 8*(lane/16)
            out[(size_t)(mbase + m) * 128 + nt * 16 + lane16] = acc[p] + bias;
        }
    }
}

// ---------------- edge messages: out[dst] += W_rel[r] @ x[src] + b_rel[r] ----
// grid = (EDGE_CHUNKS, R_TOT), 4 waves/block. Whole W_rel[r] cached in LDS
// once per block (async b128). Block loops over its edge tiles; per tile the
// 4 waves split the 8 n-tiles. Invalid (padding) rows read x row 0 and are
// dropped at scatter (independent WMMA output rows make this safe).

__global__ __launch_bounds__(128)
void edge_gemm_k(const float* __restrict__ x,
                 const float* __restrict__ Wr_all,
                 const float* __restrict__ br_all,
                 const int*   __restrict__ src,
                 const int*   __restrict__ dst,
                 const int*   __restrict__ offsets,
                 const int*   __restrict__ perm,
                 float* __restrict__ out) {
    extern __shared__ __align__(16) float smem[];
    float* lw   = smem;                         // 128 * LDS_STRIDE
    float* lx   = smem + 128 * LDS_STRIDE;      // 16  * LDS_STRIDE (shared tile)
    int*   srow = (int*)(lx + 16 * LDS_STRIDE); // 16
    int*   drow = srow + 16;                    // 16

    const int r     = blockIdx.y;
    const int start = offsets[r];
    const int nr    = offsets[r + 1] - start;
    const int Tr    = (nr + 15) >> 4;
    if ((int)blockIdx.x >= Tr) return;          // uniform exit, EXEC stays full

    const int tid    = threadIdx.x;
    const int wave   = tid >> 5;
    const int lane   = tid & 31;
    const int lane16 = lane & 15;
    const int half   = lane >> 4;

    // stage full W_rel[r] into LDS: wave w stages rows [32w, 32w+32)
    const float* Wr = Wr_all + (size_t)r * (128 * 128);
    const float* br = br_all + r * 128;
#pragma unroll 1
    for (int j = 0; j < 32; ++j) {
        int row = wave * 32 + j;
        async_load_b128(Wr + row * 128 + lane * 4, &lw[row * LDS_STRIDE + lane * 4]);
    }

#pragma unroll 1
    for (int t = blockIdx.x; t < Tr; t += EDGE_CHUNKS) {
        const int base = t * 16;
        __syncthreads();                        // prior tile's lx/srow reads done
        if (tid < 16) {
            int s = 0, d = -1;                  // padding: read x row 0, drop output
            if (base + tid < nr) {
                int e = perm[start + base + tid];
                if (e < E_FWD) { s = src[e];         d = dst[e]; }
                else           { s = dst[e - E_FWD]; d = src[e - E_FWD]; }
            }
            srow[tid] = s;
            drow[tid] = d;
        }
        __syncthreads();

        // async-gather 16 x rows; wave w stages rows [4w, 4w+4)
#pragma unroll
        for (int j = 0; j < 4; ++j) {
            int row = wave * 4 + j;
            int sr  = srow[row];
            async_load_b128(x + (size_t)sr * 128 + lane * 4,
                            &lx[row * LDS_STRIDE + lane * 4]);
        }
        wait_async0();                          // covers W staging on 1st trip too
        __syncthreads();

        // wave w computes n-tiles {2w, 2w+1}
        const float* ax = &lx[lane16 * LDS_STRIDE + half * 2];
#pragma unroll 1
        for (int q = 0; q < 2; ++q) {
            const int nt = wave * 2 + q;
            const float* bw = &lw[(nt * 16 + lane16) * LDS_STRIDE + half * 2];
            v8f acc = {};
#pragma unroll
            for (int kk = 0; kk < 32; ++kk) {
                v2f a = *(const v2f*)(ax + kk * 4);
                v2f b = *(const v2f*)(bw + kk * 4);
                acc = __builtin_amdgcn_wmma_f32_16x16x4_f32(
                    false, a, false, b, (short)0, acc, false, false);
            }
            float bias = br[nt * 16 + lane16];
#pragma unroll
            for (int p = 0; p < 8; ++p) {
                int m = p + 8 * half;
                int d = drow[m];
                if (d >= 0)
                    atomicAdd(&out[(size_t)d * 128 + nt * 16 + lane16], acc[p] + bias);
            }
        }
    }
}

// ---------------- launcher ----------------

extern "C" void kernel_launch(void* const* d_in, const int* in_sizes, int n_in,
                              void* d_out, int out_size, void* d_ws, size_t ws_size,
                              hipStream_t stream) {
    const float* x   = (const float*)d_in[0];   // [N, D]
    const float* Ws  = (const float*)d_in[1];   // [D, D]
    const float* bs  = (const float*)d_in[2];   // [D]
    const float* Wr  = (const float*)d_in[3];   // [2R, D, D]
    const float* br  = (const float*)d_in[4];   // [2R, D]
    const int*   src = (const int*)d_in[5];     // [N]
    const int*   dst = (const int*)d_in[6];     // [N]
    const int*   rel = (const int*)d_in[7];     // [N]
    float* out = (float*)d_out;                 // [N, D]

    int* ws      = (int*)d_ws;
    int* counts  = ws;            // 80
    int* offsets = ws + 80;       // 81
    int* cursors = ws + 161;      // 80
    int* perm    = ws + 241;      // 16384

    zero_counts_k <<<1, 128, 0, stream>>>(counts);
    count_edges_k <<<(E_ALL + 255) / 256, 256, 0, stream>>>(rel, counts);
    scan_offsets_k<<<1, 32, 0, stream>>>(counts, offsets, cursors);
    scatter_edges_k<<<(E_ALL + 255) / 256, 256, 0, stream>>>(rel, cursors, perm);

    // self transform fully writes out (no pre-zero needed)
    const size_t self_smem = (128 * LDS_STRIDE + SELF_MTILES * 16 * LDS_STRIDE) * sizeof(float);
    self_gemm_k<<<N_TOK / (16 * SELF_MTILES), 128, self_smem, stream>>>(x, Ws, bs, out);

    const size_t edge_smem = (128 * LDS_STRIDE + 16 * LDS_STRIDE) * sizeof(float)
                           + 32 * sizeof(int);
    dim3 egrid(EDGE_CHUNKS, R_TOT);
    edge_gemm_k<<<egrid, 128, edge_smem, stream>>>(x, Wr, br, src, dst, offsets, perm, out);
}